// YOLOLoss_8581344657529
// MI455X (gfx1250) — compile-verified
//
#include <hip/hip_runtime.h>
#include <hip/hip_bf16.h>

// ---------------------------------------------------------------------------
// YOLOX SimOTA loss for gfx1250 (MI455X).
// B=16, A=3 (only anchor 0 used), H=W=128 -> N=16384, D=85, G=60, C=80.
// The onehot @ cls^T einsum runs on V_WMMA_F32_16X16X4_F32 (exact: one-hot
// rows select single f32 values, so accumulation split/order is irrelevant).
// Each wave computes TWO 16x16 output tiles with independent accumulator
// chains to hide WMMA D->C RAW latency, and hoists the one-hot A fragments.
// ---------------------------------------------------------------------------

#define NB     16
#define NANCH  16384
#define PD     85
#define NCLS   80
#define GTS    60
#define GPAD   64
#define EPSF   1e-8f

typedef float v2f __attribute__((ext_vector_type(2)));
typedef float v8f __attribute__((ext_vector_type(8)));

__device__ __forceinline__ float softplusf(float x) {
  // stable log1p(exp(x)) == logaddexp(x, 0)
  return fmaxf(x, 0.0f) + log1pf(expf(-fabsf(x)));
}

__device__ __forceinline__ float iou_box(float ax1, float ay1, float ax2, float ay2,
                                         float bx1, float by1, float bx2, float by2) {
  float w = fmaxf(fminf(ax2, bx2) - fmaxf(ax1, bx1), 0.0f);
  float h = fmaxf(fminf(ay2, by2) - fmaxf(ay1, by1), 0.0f);
  float inter = w * h;
  float aa = fmaxf(ax2 - ax1, 0.0f) * fmaxf(ay2 - ay1, 0.0f);
  float ab = fmaxf(bx2 - bx1, 0.0f) * fmaxf(by2 - by1, 0.0f);
  return inter / (aa + ab - inter + EPSF);
}

// ---------------------------------------------------------------------------
// Kernel A: per-anchor sum of softplus over the 80 class logits (all images).
// ---------------------------------------------------------------------------
__global__ void sp_sum_kernel(const float* __restrict__ preds, float* __restrict__ sp) {
  int idx = blockIdx.x * blockDim.x + threadIdx.x;
  if (idx >= NB * NANCH) return;
  int b = idx >> 14;
  int n = idx & (NANCH - 1);
  const float* p = preds + ((size_t)b * 3 * NANCH + n) * PD + 5;
  float s = 0.0f;
  #pragma unroll 4
  for (int c = 0; c < NCLS; ++c) s += softplusf(p[c]);
  sp[idx] = s;
}

// ---------------------------------------------------------------------------
// Kernel W: clsdot[g, n] = cls_pred[n, label_g]  ==  (onehot @ cls^T)[g, n]
// One wave32 per pair of 16x16 output tiles (same g-tile, adjacent n-tiles).
// K=80 via 20 x V_WMMA_F32_16X16X4_F32 per tile, two interleaved chains.
// A 16x4 f32 layout: lanes 0-15 hold (K=0,K=1), lanes 16-31 hold (K=2,K=3);
// B 4x16 mirrors A; C/D: VGPR r holds rows M=r (lanes 0-15) and M=r+8.
// ---------------------------------------------------------------------------
__global__ void wmma_clsdot_kernel(const float* __restrict__ pred,   // image base (anchor 0)
                                   const int*   __restrict__ labels, // 60 labels
                                   float*       __restrict__ clsdot) // 64 x 16384
{
  const int n0   = blockIdx.x << 5;   // two adjacent 16-wide n tiles
  const int g0   = blockIdx.y << 4;
  const int lane = threadIdx.x;       // 0..31, block is exactly one wave
  const int half = lane >> 4;         // 0: K pair (0,1), 1: K pair (2,3)
  const int l15  = lane & 15;

  const int g   = g0 + l15;
  const int lab = (g < GTS) ? labels[g] : -1;   // padded rows -> zero row

  // Hoist the 20 one-hot A fragments (depend only on labels).
  v2f afrag[20];
  #pragma unroll
  for (int s = 0; s < 20; ++s) {
    const int ka = s * 4 + half * 2;
    afrag[s].x = (lab == ka)     ? 1.0f : 0.0f;
    afrag[s].y = (lab == ka + 1) ? 1.0f : 0.0f;
  }

  const float* pn0 = pred + (size_t)(n0 + l15) * PD + 5;        // tile 0 column
  const float* pn1 = pred + (size_t)(n0 + 16 + l15) * PD + 5;   // tile 1 column

  v8f acc0 = {};
  v8f acc1 = {};
  #pragma unroll
  for (int s = 0; s < 20; ++s) {
    const int ka = s * 4 + half * 2;
    v2f b0, b1;
    b0.x = pn0[ka]; b0.y = pn0[ka + 1];
    b1.x = pn1[ka]; b1.y = pn1[ka + 1];
    // Two independent accumulator chains: alternating issues hide the
    // WMMA D->C RAW latency of a single serialized chain.
    acc0 = __builtin_amdgcn_wmma_f32_16x16x4_f32(
        false, afrag[s], false, b0, (short)0, acc0, false, false);
    acc1 = __builtin_amdgcn_wmma_f32_16x16x4_f32(
        false, afrag[s], false, b1, (short)0, acc1, false, false);
  }

  #pragma unroll
  for (int r = 0; r < 8; ++r) {
    int row = g0 + r + half * 8;
    clsdot[(size_t)row * NANCH + n0 + l15]      = acc0[r];
    clsdot[(size_t)row * NANCH + n0 + 16 + l15] = acc1[r];
  }
}

// ---------------------------------------------------------------------------
// Kernel B: per gt row -> dynamic_k (top-10 IoU sum) and cost threshold
// (k-th smallest cost). One 256-thread block per gt.
// ---------------------------------------------------------------------------
__global__ void topk_kernel(const float* __restrict__ gtb,    // 60 x 4
                            const float* __restrict__ pred,   // image base
                            const float* __restrict__ clsdot, // 64 x N
                            const float* __restrict__ sp,     // N
                            float*       __restrict__ thr)    // 64
{
  const int g   = blockIdx.x;
  const int tid = threadIdx.x;

  __shared__ float sgt[4];
  if (tid < 4) sgt[tid] = gtb[g * 4 + tid];
  __syncthreads();
  const float ax1 = sgt[0], ay1 = sgt[1], ax2 = sgt[2], ay2 = sgt[3];

  float ti[10], tc[10];
  #pragma unroll
  for (int i = 0; i < 10; ++i) { ti[i] = -1e30f; tc[i] = 1e30f; }

  const float* crow = clsdot + (size_t)g * NANCH;
  for (int n = tid; n < NANCH; n += 256) {
    const float* p = pred + (size_t)n * PD;
    // Pull the next strip toward the WGP while we do the IoU/log math.
    if (n + 256 < NANCH) __builtin_prefetch(pred + (size_t)(n + 256) * PD, 0, 0);
    float iou  = iou_box(ax1, ay1, ax2, ay2, p[0], p[1], p[2], p[3]);
    float cost = sp[n] - crow[n] + 3.0f * (-logf(iou + EPSF));
    if (iou > ti[9]) {
      int j = 9;
      while (j > 0 && ti[j - 1] < iou) { ti[j] = ti[j - 1]; --j; }
      ti[j] = iou;
    }
    if (cost < tc[9]) {
      int j = 9;
      while (j > 0 && tc[j - 1] > cost) { tc[j] = tc[j - 1]; --j; }
      tc[j] = cost;
    }
  }

  __shared__ float si[256 * 10];
  __shared__ float sc[256 * 10];
  #pragma unroll
  for (int i = 0; i < 10; ++i) { si[tid * 10 + i] = ti[i]; sc[tid * 10 + i] = tc[i]; }
  __syncthreads();

  if (tid == 0) {
    float bi[10], bc[10];
    #pragma unroll
    for (int i = 0; i < 10; ++i) { bi[i] = -1e30f; bc[i] = 1e30f; }
    for (int t = 0; t < 256 * 10; ++t) {
      float v = si[t];
      if (v > bi[9]) {
        int j = 9;
        while (j > 0 && bi[j - 1] < v) { bi[j] = bi[j - 1]; --j; }
        bi[j] = v;
      }
      float c = sc[t];
      if (c < bc[9]) {
        int j = 9;
        while (j > 0 && bc[j - 1] > c) { bc[j] = bc[j - 1]; --j; }
        bc[j] = c;
      }
    }
    float s = 0.0f;
    #pragma unroll
    for (int i = 0; i < 10; ++i) s += bi[i];
    s = fmaxf(s, 1.0f);          // clip(sum, 1.0)
    int k = (int)s;              // trunc, in [1, 10]
    if (k > 10) k = 10;
    thr[g] = bc[k - 1];          // k-th smallest cost <=> rank < k
  }
}

// ---------------------------------------------------------------------------
// Kernel C: per-anchor matching resolution + loss accumulation (one image).
// ---------------------------------------------------------------------------
__global__ void match_kernel(const float* __restrict__ gtb,
                             const int*   __restrict__ glab,
                             const float* __restrict__ pred,
                             const float* __restrict__ clsdot,
                             const float* __restrict__ sp,
                             const float* __restrict__ thr,
                             float*       __restrict__ accum) // [4]: nfg, box, obj, cls
{
  __shared__ float sb[GTS * 4];
  __shared__ int   sl[GTS];
  __shared__ float st[GTS];
  const int tid = threadIdx.x;
  for (int i = tid; i < GTS * 4; i += 256) sb[i] = gtb[i];
  for (int i = tid; i < GTS; i += 256) { sl[i] = glab[i]; st[i] = thr[i]; }
  __syncthreads();

  const int n = blockIdx.x * 256 + tid;   // grid is exactly 64 x 256
  const float* p = pred + (size_t)n * PD;
  const float bx1 = p[0], by1 = p[1], bx2 = p[2], by2 = p[3], obj = p[4];
  const float sps = sp[n];

  unsigned long long mask = 0ull;
  float bestc = 1e38f;
  int   bestg = 0;
  for (int g = 0; g < GTS; ++g) {
    // Next clsdot row for this anchor is a strided read; prefetch it.
    if (g + 1 < GTS) __builtin_prefetch(clsdot + (size_t)(g + 1) * NANCH + n, 0, 0);
    float iou  = iou_box(sb[g * 4], sb[g * 4 + 1], sb[g * 4 + 2], sb[g * 4 + 3],
                         bx1, by1, bx2, by2);
    float cost = sps - clsdot[(size_t)g * NANCH + n] + 3.0f * (-logf(iou + EPSF));
    if (cost < bestc) { bestc = cost; bestg = g; }          // first-min tie rule
    if (cost <= st[g]) mask |= (1ull << g);
  }

  int  cnt = __builtin_popcountll(mask);
  bool fg  = false;
  int  mg  = 0;
  if (cnt == 1)      { fg = true; mg = __builtin_ctzll(mask); }
  else if (cnt > 1)  { fg = ((mask >> bestg) & 1ull) != 0ull; mg = bestg; }

  float f = 0.0f, bl = 0.0f, ol = 0.0f, cl = 0.0f;
  if (fg) {
    f = 1.0f;
    const float gx1 = sb[mg * 4], gy1 = sb[mg * 4 + 1], gx2 = sb[mg * 4 + 2], gy2 = sb[mg * 4 + 3];
    const float pb[4] = {bx1, by1, bx2, by2};
    const float gb[4] = {gx1, gy1, gx2, gy2};
    #pragma unroll
    for (int c = 0; c < 4; ++c) {
      float d  = pb[c] - gb[c];
      float ad = fabsf(d);
      bl += (ad < 1.0f) ? (0.5f * d * d) : (ad - 0.5f);
    }
    float iou_t = iou_box(bx1, by1, bx2, by2, gx1, gy1, gx2, gy2);
    float obj_s = 1.0f / (1.0f + expf(-obj));
    ol = softplusf(obj_s) - obj_s * iou_t;              // reference's bce(sigmoid(x), iou)
    cl = sps - clsdot[(size_t)mg * NANCH + n];          // sum_c bce(cls, onehot(label))
  }

  __shared__ float red[4][256];
  red[0][tid] = f; red[1][tid] = bl; red[2][tid] = ol; red[3][tid] = cl;
  __syncthreads();
  for (int s = 128; s > 0; s >>= 1) {
    if (tid < s) {
      red[0][tid] += red[0][tid + s];
      red[1][tid] += red[1][tid + s];
      red[2][tid] += red[2][tid + s];
      red[3][tid] += red[3][tid + s];
    }
    __syncthreads();
  }
  if (tid == 0) {
    atomicAdd(&accum[0], red[0][0]);
    atomicAdd(&accum[1], red[1][0]);
    atomicAdd(&accum[2], red[2][0]);
    atomicAdd(&accum[3], red[3][0]);
  }
}

// ---------------------------------------------------------------------------
// Kernel D: per-image normalization and final 4 scalars.
// ---------------------------------------------------------------------------
__global__ void finalize_kernel(const float* __restrict__ accum, float* __restrict__ out) {
  if (threadIdx.x != 0 || blockIdx.x != 0) return;
  float tb = 0.0f, to = 0.0f, tc = 0.0f;
  for (int b = 0; b < NB; ++b) {
    float nfg = fmaxf(accum[b * 4 + 0], 1.0f);
    tb += accum[b * 4 + 1] / (nfg * 4.0f);
    to += accum[b * 4 + 2] / nfg;
    tc += accum[b * 4 + 3] / (nfg * (float)NCLS);
  }
  const float invB = 1.0f / (float)NB;
  out[0] = (5.0f * tb + to + tc) * invB;
  out[1] = tb * invB;
  out[2] = to * invB;
  out[3] = tc * invB;
}

// ---------------------------------------------------------------------------
// Host orchestration. Workspace layout (bytes):
//   [0)                 sp_sum   : NB*NANCH floats (1 MB)
//   [SP_B)              clsdot   : GPAD*NANCH floats (4 MB, reused per image)
//   [SP_B+CD_B)         thr      : 64 floats (reused per image)
//   [.. +256)           accum    : NB*4 floats (zeroed each launch)
// ---------------------------------------------------------------------------
extern "C" void kernel_launch(void* const* d_in, const int* in_sizes, int n_in,
                              void* d_out, int out_size, void* d_ws, size_t ws_size,
                              hipStream_t stream) {
  const float* preds     = (const float*)d_in[0];
  const float* gt_boxes  = (const float*)d_in[1];
  const int*   gt_labels = (const int*)d_in[2];
  float* out = (float*)d_out;

  char* ws = (char*)d_ws;
  const size_t SP_B = (size_t)NB * NANCH * sizeof(float);
  const size_t CD_B = (size_t)GPAD * NANCH * sizeof(float);
  float* sp     = (float*)ws;
  float* clsdot = (float*)(ws + SP_B);
  float* thr    = (float*)(ws + SP_B + CD_B);
  float* accum  = thr + GPAD;

  hipMemsetAsync(accum, 0, NB * 4 * sizeof(float), stream);

  sp_sum_kernel<<<(NB * NANCH + 255) / 256, 256, 0, stream>>>(preds, sp);

  for (int b = 0; b < NB; ++b) {
    const float* pimg = preds + (size_t)b * 3 * NANCH * PD;   // anchor 0 of image b
    const float* gbb  = gt_boxes + (size_t)b * GTS * 4;
    const int*   glb  = gt_labels + (size_t)b * GTS;

    dim3 gw(NANCH / 32, GPAD / 16);   // each wave: two adjacent n tiles
    wmma_clsdot_kernel<<<gw, 32, 0, stream>>>(pimg, glb, clsdot);
    topk_kernel<<<GTS, 256, 0, stream>>>(gbb, pimg, clsdot, sp + (size_t)b * NANCH, thr);
    match_kernel<<<NANCH / 256, 256, 0, stream>>>(gbb, glb, pimg, clsdot,
                                                  sp + (size_t)b * NANCH, thr,
                                                  accum + b * 4);
  }

  finalize_kernel<<<1, 32, 0, stream>>>(accum, out);
}